// Sparse_En_MHAtt_33689723470279
// MI455X (gfx1250) — compile-verified
//
#include <hip/hip_runtime.h>
#include <hip/hip_bf16.h>
#include <stddef.h>

// CDNA5 / gfx1250, wave32. Full-f32 WMMA path (V_WMMA_F32_16X16X4_F32):
// exact precision for the sparsity-threshold comparison, matrix pipe for all GEMMs.
// GEMM staging uses the CDNA5 async global->LDS path (ASYNCcnt) with LDS
// double-buffering when the builtin is available.

typedef __attribute__((ext_vector_type(2))) float v2f;
typedef __attribute__((ext_vector_type(8))) float v8f;
typedef __attribute__((ext_vector_type(4))) int   v4i;

#define WMMA_F32(A, B, C) \
    __builtin_amdgcn_wmma_f32_16x16x4_f32(false, (A), false, (B), (short)0, (C), false, false)

#if defined(__has_builtin)
#if __has_builtin(__builtin_amdgcn_global_load_async_to_lds_b128)
#define USE_ASYNC_LDS 1
#endif
#endif

#ifdef USE_ASYNC_LDS
// Probe-derived signature: (int4 __global* src, int4 __shared* dst, imm offset, imm cpol)
typedef __attribute__((address_space(1))) void as1_void;
typedef __attribute__((address_space(3))) void as3_void;
typedef __attribute__((address_space(1))) v4i  as1_v4i;
typedef __attribute__((address_space(3))) v4i  as3_v4i;
#define ASYNC_CP_B128(gsrc, ldst)                                   \
    __builtin_amdgcn_global_load_async_to_lds_b128(                 \
        (as1_v4i*)(as1_void*)(void*)(gsrc),                         \
        (as3_v4i*)(as3_void*)(ldst), 0, 0)

__device__ __forceinline__ void wait_async0() {
#if __has_builtin(__builtin_amdgcn_s_wait_asynccnt)
    __builtin_amdgcn_s_wait_asynccnt(0);
#else
    asm volatile("s_wait_asynccnt 0x0" ::: "memory");
#endif
}
#endif

// ---------------------------------------------------------------------------
// GEMM: Out[m, e] = sum_k X[m,k] * W[e,k] + bias[e]
// M = 4096, N = K = 1024.  64x64 block tile, 256 threads (8 waves).
// Wave layout: ntile = wave&3 (16 cols), two m-tiles (wave>>2)*2, (wave>>2)*2+1.
// headsplit=1 -> write [B,H,S,64]; headsplit=0 -> write [M,N] row-major.
// ---------------------------------------------------------------------------
__global__ __launch_bounds__(256) void Gemm_XWt(const float* __restrict__ X,
                                                const float* __restrict__ W,
                                                const float* __restrict__ bias,
                                                float* __restrict__ Out,
                                                int headsplit) {
    const int tid   = threadIdx.x;
    const int lane  = tid & 31;
    const int wave  = tid >> 5;
    const int ntile = wave & 3;
    const int mt0   = (wave >> 2) << 1;       // 0 or 2
    const int m0    = blockIdx.y * 64;
    const int n0    = blockIdx.x * 64;

    const int l16 = lane & 15;
    const int khi = (lane >> 4) * 2;          // k base per half-wave (A/B 16x4 layout)

    const int lrow = tid >> 2;                // 0..63
    const int lcol = (tid & 3) * 8;           // 0,8,16,24

    v8f acc0 = {};
    v8f acc1 = {};

#ifdef USE_ASYNC_LDS
    // Double-buffered LDS chunks filled by async global->LDS copies (ASYNCcnt).
    __shared__ float As[2][64 * 36];
    __shared__ float Bs[2][64 * 36];

    // Prologue: kick off chunk 0 into buffer 0.
    {
        const float* ga = &X[(size_t)(m0 + lrow) * 1024 + lcol];
        const float* gb = &W[(size_t)(n0 + lrow) * 1024 + lcol];
        ASYNC_CP_B128(ga,     &As[0][lrow * 36 + lcol]);
        ASYNC_CP_B128(ga + 4, &As[0][lrow * 36 + lcol + 4]);
        ASYNC_CP_B128(gb,     &Bs[0][lrow * 36 + lcol]);
        ASYNC_CP_B128(gb + 4, &Bs[0][lrow * 36 + lcol + 4]);
    }

    for (int c = 0; c < 32; ++c) {
        const int cur = c & 1;
        wait_async0();        // my copies into buf[cur] are in LDS
        __syncthreads();      // everyone's copies visible; prior compute done

        if (c + 1 < 32) {     // overlap: fetch next chunk into the other buffer
            const int k1 = (c + 1) * 32;
            const float* ga = &X[(size_t)(m0 + lrow) * 1024 + k1 + lcol];
            const float* gb = &W[(size_t)(n0 + lrow) * 1024 + k1 + lcol];
            ASYNC_CP_B128(ga,     &As[cur ^ 1][lrow * 36 + lcol]);
            ASYNC_CP_B128(ga + 4, &As[cur ^ 1][lrow * 36 + lcol + 4]);
            ASYNC_CP_B128(gb,     &Bs[cur ^ 1][lrow * 36 + lcol]);
            ASYNC_CP_B128(gb + 4, &Bs[cur ^ 1][lrow * 36 + lcol + 4]);
        }

#pragma unroll
        for (int ks = 0; ks < 8; ++ks) {
            const int kk = khi + 4 * ks;
            v2f bf  = *(const v2f*)&Bs[cur][(ntile * 16 + l16) * 36 + kk];
            v2f af0 = *(const v2f*)&As[cur][((mt0    ) * 16 + l16) * 36 + kk];
            v2f af1 = *(const v2f*)&As[cur][((mt0 + 1) * 16 + l16) * 36 + kk];
            acc0 = WMMA_F32(af0, bf, acc0);
            acc1 = WMMA_F32(af1, bf, acc1);
        }
    }
#else
    // Fallback: register-staged global loads + ds_store.
    __shared__ float As[64 * 36];
    __shared__ float Bs[64 * 36];

    for (int k0 = 0; k0 < 1024; k0 += 32) {
        const float* ga = &X[(size_t)(m0 + lrow) * 1024 + k0 + lcol];
        const float* gb = &W[(size_t)(n0 + lrow) * 1024 + k0 + lcol];
        float4 a0 = *(const float4*)(ga);
        float4 a1 = *(const float4*)(ga + 4);
        float4 b0 = *(const float4*)(gb);
        float4 b1 = *(const float4*)(gb + 4);
        __syncthreads();  // previous iteration's LDS reads done
        *(float4*)&As[lrow * 36 + lcol]     = a0;
        *(float4*)&As[lrow * 36 + lcol + 4] = a1;
        *(float4*)&Bs[lrow * 36 + lcol]     = b0;
        *(float4*)&Bs[lrow * 36 + lcol + 4] = b1;
        __syncthreads();

#pragma unroll
        for (int ks = 0; ks < 8; ++ks) {
            const int kk = khi + 4 * ks;
            v2f bf  = *(const v2f*)&Bs[(ntile * 16 + l16) * 36 + kk];
            v2f af0 = *(const v2f*)&As[((mt0    ) * 16 + l16) * 36 + kk];
            v2f af1 = *(const v2f*)&As[((mt0 + 1) * 16 + l16) * 36 + kk];
            acc0 = WMMA_F32(af0, bf, acc0);
            acc1 = WMMA_F32(af1, bf, acc1);
        }
    }
#endif

    // Epilogue: D layout reg r -> m = r + 8*(lane>>4), n = lane&15
    const int ng = n0 + ntile * 16 + l16;
    const float bv = bias[ng];
#pragma unroll
    for (int r = 0; r < 8; ++r) {
        const int mr  = r + 8 * (lane >> 4);
        const int mg0 = m0 + mt0 * 16 + mr;
        const int mg1 = mg0 + 16;
        const float o0 = acc0[r] + bv;
        const float o1 = acc1[r] + bv;
        if (headsplit) {
            const int h = ng >> 6, d = ng & 63;
            const int b0i = mg0 >> 10, s0 = mg0 & 1023;
            const int b1i = mg1 >> 10, s1 = mg1 & 1023;
            Out[((((size_t)b0i * 16 + h) << 10) + s0) * 64 + d] = o0;
            Out[((((size_t)b1i * 16 + h) << 10) + s1) * 64 + d] = o1;
        } else {
            Out[(size_t)mg0 * 1024 + ng] = o0;
            Out[(size_t)mg1 * 1024 + ng] = o1;
        }
    }
}

// ---------------------------------------------------------------------------
// Attention: one block per (b*H+h, 16-query tile).  Full 16x1024 f32 score
// row-block in LDS; two-pass softmax with sparsity mask done per-row inside a
// single wave (16 lanes per row, shfl_xor width-16 reductions); ctx via WMMA.
// ---------------------------------------------------------------------------
__global__ __launch_bounds__(256) void Attn16(const float* __restrict__ Qh,
                                              const float* __restrict__ Kh,
                                              const float* __restrict__ Vh,
                                              float* __restrict__ Ctx) {
    __shared__ float sc[16 * 1024];           // 64 KB score tile

    const int bh = blockIdx.y;                // b*16 + h
    const int qt = blockIdx.x;                // query tile (16 rows)
    const float* Qb = Qh + (size_t)bh * 65536;
    const float* Kb = Kh + (size_t)bh * 65536;
    const float* Vb = Vh + (size_t)bh * 65536;

    const int tid  = threadIdx.x;
    const int lane = tid & 31;
    const int wave = tid >> 5;
    const int l16  = lane & 15;
    const int khi  = (lane >> 4) * 2;

    // Preload Q fragments for all 16 k-steps (rows qt*16 + l16, K = 64).
    v2f qa[16];
#pragma unroll
    for (int ks = 0; ks < 16; ++ks)
        qa[ks] = *(const v2f*)&Qb[(size_t)(qt * 16 + l16) * 64 + khi + 4 * ks];

    // ---- Phase A: scores = (Q K^T) / sqrt(64); each wave owns 8 key tiles ---
    for (int t = 0; t < 8; ++t) {
        const int keyBase = (wave * 8 + t) * 16;
        v8f acc = {};
#pragma unroll
        for (int ks = 0; ks < 16; ++ks) {
            v2f bf = *(const v2f*)&Kb[(size_t)(keyBase + l16) * 64 + khi + 4 * ks];
            acc = WMMA_F32(qa[ks], bf, acc);
        }
#pragma unroll
        for (int r = 0; r < 8; ++r) {
            const int m = r + 8 * (lane >> 4);
            sc[m * 1024 + keyBase + l16] = acc[r] * 0.125f;  // 1/sqrt(dk)
        }
    }
    __syncthreads();

    // ---- Two-pass masked softmax.  Row m owned by threads [16m,16m+16). ----
    {
        const int row = tid >> 4;
        const int sub = tid & 15;
        float* srow = &sc[row * 1024];

        // pass 1: softmax statistics of raw scores
        float mx = -3.402823e38f;
        for (int j = sub; j < 1024; j += 16) mx = fmaxf(mx, srow[j]);
        for (int off = 8; off > 0; off >>= 1) mx = fmaxf(mx, __shfl_xor(mx, off, 16));
        float sum = 0.f;
        for (int j = sub; j < 1024; j += 16) sum += __expf(srow[j] - mx);
        for (int off = 8; off > 0; off >>= 1) sum += __shfl_xor(sum, off, 16);
        const float inv = 1.f / sum;
        // mask: att1 < 0.09 -> score := -1e9, else keep raw score
        for (int j = sub; j < 1024; j += 16) {
            const float s  = srow[j];
            const float a1 = __expf(s - mx) * inv;
            srow[j] = (a1 < 0.09f) ? -1.0e9f : s;
        }
        // pass 2: softmax of masked scores (all-masked row -> uniform, as ref)
        float mx2 = -3.402823e38f;
        for (int j = sub; j < 1024; j += 16) mx2 = fmaxf(mx2, srow[j]);
        for (int off = 8; off > 0; off >>= 1) mx2 = fmaxf(mx2, __shfl_xor(mx2, off, 16));
        float sum2 = 0.f;
        for (int j = sub; j < 1024; j += 16) sum2 += __expf(srow[j] - mx2);
        for (int off = 8; off > 0; off >>= 1) sum2 += __shfl_xor(sum2, off, 16);
        const float inv2 = 1.f / sum2;
        for (int j = sub; j < 1024; j += 16)
            srow[j] = __expf(srow[j] - mx2) * inv2;  // att2
    }
    __syncthreads();

    // ---- Phase C: ctx = att2 @ V. Waves 0..3 each own one 16-col tile. ----
    if (wave < 4) {
        const int nb = wave * 16;
        v8f acc = {};
        for (int ks = 0; ks < 256; ++ks) {
            const int kk = khi + 4 * ks;
            v2f af = *(const v2f*)&sc[l16 * 1024 + kk];  // att2 fragment from LDS
            v2f bf;
            bf.x = Vb[(size_t)(kk    ) * 64 + nb + l16];
            bf.y = Vb[(size_t)(kk + 1) * 64 + nb + l16];
            acc = WMMA_F32(af, bf, acc);
        }
        const int b = bh >> 4, h = bh & 15;
#pragma unroll
        for (int r = 0; r < 8; ++r) {
            const int m = r + 8 * (lane >> 4);
            const int s = qt * 16 + m;
            Ctx[((size_t)(b * 1024 + s)) * 1024 + h * 64 + nb + l16] = acc[r];
        }
    }
}

// ---------------------------------------------------------------------------
extern "C" void kernel_launch(void* const* d_in, const int* in_sizes, int n_in,
                              void* d_out, int out_size, void* d_ws, size_t ws_size,
                              hipStream_t stream) {
    (void)in_sizes; (void)n_in; (void)out_size; (void)ws_size;

    const float* q  = (const float*)d_in[0];
    const float* k  = (const float*)d_in[1];
    const float* v  = (const float*)d_in[2];
    const float* Wq = (const float*)d_in[3];
    const float* bq = (const float*)d_in[4];
    const float* Wk = (const float*)d_in[5];
    const float* bk = (const float*)d_in[6];
    const float* Wv = (const float*)d_in[7];
    const float* bv = (const float*)d_in[8];
    const float* Wo = (const float*)d_in[9];
    const float* bo = (const float*)d_in[10];

    // Workspace: Qh, Kh, Vh [B,H,S,64] + Ctx [B,S,D], 16 MB each (64 MB total)
    float* Qh  = (float*)d_ws;
    float* Kh  = Qh + (size_t)4 * 16 * 1024 * 64;
    float* Vh  = Kh + (size_t)4 * 16 * 1024 * 64;
    float* Ctx = Vh + (size_t)4 * 16 * 1024 * 64;

    dim3 gGemm(16, 64);   // x: N/64, y: M/64  (M=4096, N=1024)
    dim3 block(256);

    Gemm_XWt<<<gGemm, block, 0, stream>>>(q, Wq, bq, Qh, 1);
    Gemm_XWt<<<gGemm, block, 0, stream>>>(k, Wk, bk, Kh, 1);
    Gemm_XWt<<<gGemm, block, 0, stream>>>(v, Wv, bv, Vh, 1);

    dim3 gAttn(64, 64);   // x: S/16 query tiles, y: B*H
    Attn16<<<gAttn, block, 0, stream>>>(Qh, Kh, Vh, Ctx);

    Gemm_XWt<<<gGemm, block, 0, stream>>>(Ctx, Wo, bo, (float*)d_out, 0);
}